// GcnNet_39797166965052
// MI455X (gfx1250) — compile-verified
//
#include <hip/hip_runtime.h>
#include <hip/hip_bf16.h>

// ---------------- problem constants (match reference) ----------------
#define N_GRAPHS   8
#define NODES_PER_G 4096
#define NN         (N_GRAPHS * NODES_PER_G)   // 32768 nodes
#define EE         (NN * 16)                  // 524288 edges
#define F_IN       128
#define HID        64
#define FC_HID     128
#define OUT_DIM    2
#define SLOPE      0.01f

typedef float v2f __attribute__((ext_vector_type(2)));
typedef float v8f __attribute__((ext_vector_type(8)));

__device__ __forceinline__ float leaky(float x) {
    return x >= 0.0f ? x : SLOPE * x;
}

__device__ __forceinline__ void atomic_add_f32(float* p, float v) {
    // agent-scope relaxed fp32 add -> global_atomic_add_f32 (non-returning)
    __hip_atomic_fetch_add(p, v, __ATOMIC_RELAXED, __HIP_MEMORY_SCOPE_AGENT);
}

// ---------------- workspace layout (floats) ----------------
// [0, NN)                       : deg -> dinv (in place)
// [NN, NN + NN*HID)             : agg1 -> h1 (in place)
// [NN + NN*HID, NN*66)          : agg2
// [NN*66, NN*130)               : xw1
// [NN*130, NN*131)              : xw2 -> h2 (in place)
#define OFF_DINV  0
#define OFF_AGG1  (NN)
#define OFF_AGG2  (NN + NN * HID)
#define OFF_XW1   (NN * 66)
#define OFF_XW2   (NN * 130)
#define ZERO_CNT  (NN * 66)   // dinv + agg1 + agg2 are contiguous

// ---------------- kernels ----------------

__global__ void zero_ws(float* __restrict__ ws) {
    int i = blockIdx.x * blockDim.x + threadIdx.x;
    if (i < ZERO_CNT) ws[i] = 0.0f;
}

__global__ void deg_scatter(const int* __restrict__ dst,
                            const float* __restrict__ edge_attr,
                            float* __restrict__ deg) {
    int e = blockIdx.x * blockDim.x + threadIdx.x;
    if (e < EE) atomic_add_f32(&deg[dst[e]], edge_attr[2 * e]);
}

__global__ void dinv_finalize(float* __restrict__ deg) {
    int i = blockIdx.x * blockDim.x + threadIdx.x;
    if (i < NN) {
        float d = deg[i] + 1.0f;          // self-loop weight 1
        deg[i] = __frsqrt_rn(d);          // d >= 1 always
    }
}

// xw1 = x @ W1 using V_WMMA_F32_16X16X4_F32.
// One wave -> 16 rows x 64 cols (4 col-tiles share the A fragment).
// A (16x4 f32): lane<16 holds M=lane, K=k,k+1 ; lane>=16 holds M=lane-16, K=k+2,k+3
// B (4x16 f32): lane<16 holds N=lane, K=k,k+1 ; lane>=16 holds N=lane-16, K=k+2,k+3
// C (16x16 f32): VGPR r: lanes0-15 -> (M=r, N=lane); lanes16-31 -> (M=r+8, N=lane-16)
__global__ void gemm1_wmma(const float* __restrict__ x,
                           const float* __restrict__ W1,
                           float* __restrict__ xw1) {
    int wave = (blockIdx.x * blockDim.x + threadIdx.x) >> 5;
    int lane = threadIdx.x & 31;
    int row0 = wave * 16;
    if (row0 >= NN) return;

    int m     = lane & 15;            // M (for A) / N (for B) sub-index
    int khalf = (lane >> 4) << 1;     // 0 for lanes 0-15, 2 for lanes 16-31

    v8f c0 = {}, c1 = {}, c2 = {}, c3 = {};
    const float* xrow = x + (size_t)(row0 + m) * F_IN;

    #pragma unroll 4
    for (int k = 0; k < F_IN; k += 4) {
        // A fragment: two consecutive K values for this lane's row
        float2 av = *(const float2*)(xrow + k + khalf);
        v2f a; a.x = av.x; a.y = av.y;

        const float* wk0 = W1 + (size_t)(k + khalf) * HID;       // row k+khalf
        const float* wk1 = W1 + (size_t)(k + khalf + 1) * HID;   // row k+khalf+1

        v2f b0, b1, b2, b3;
        b0.x = wk0[m +  0]; b0.y = wk1[m +  0];
        b1.x = wk0[m + 16]; b1.y = wk1[m + 16];
        b2.x = wk0[m + 32]; b2.y = wk1[m + 32];
        b3.x = wk0[m + 48]; b3.y = wk1[m + 48];

        c0 = __builtin_amdgcn_wmma_f32_16x16x4_f32(false, a, false, b0, (short)0, c0, false, false);
        c1 = __builtin_amdgcn_wmma_f32_16x16x4_f32(false, a, false, b1, (short)0, c1, false, false);
        c2 = __builtin_amdgcn_wmma_f32_16x16x4_f32(false, a, false, b2, (short)0, c2, false, false);
        c3 = __builtin_amdgcn_wmma_f32_16x16x4_f32(false, a, false, b3, (short)0, c3, false, false);
    }

    int rbase = (lane >> 4) << 3;     // 0 or 8
    int nlo   = lane & 15;
    #pragma unroll
    for (int r = 0; r < 8; ++r) {
        size_t row = (size_t)(row0 + rbase + r);
        float* o = xw1 + row * HID + nlo;
        o[ 0] = c0[r];
        o[16] = c1[r];
        o[32] = c2[r];
        o[48] = c3[r];
    }
}

// agg1[dst] += xw1[src] * norm   (16 threads per edge, float4 gathers)
__global__ void edge_scatter1(const int* __restrict__ src,
                              const int* __restrict__ dst,
                              const float* __restrict__ edge_attr,
                              const float* __restrict__ dinv,
                              const float* __restrict__ xw1,
                              float* __restrict__ agg1) {
    long long idx = (long long)blockIdx.x * blockDim.x + threadIdx.x;
    if (idx >= (long long)EE * 16) return;
    int e    = (int)(idx >> 4);
    int f0   = ((int)idx & 15) * 4;
    int s    = src[e], d = dst[e];
    float nw = dinv[s] * edge_attr[2 * e] * dinv[d];
    float4 v = *(const float4*)(xw1 + (size_t)s * HID + f0);
    float* o = agg1 + (size_t)d * HID + f0;
    atomic_add_f32(o + 0, v.x * nw);
    atomic_add_f32(o + 1, v.y * nw);
    atomic_add_f32(o + 2, v.z * nw);
    atomic_add_f32(o + 3, v.w * nw);
}

// h1 = leaky(agg1 + xw1 * dinv^2 + b1), written in place over agg1
__global__ void combine1(float* __restrict__ agg1,
                         const float* __restrict__ xw1,
                         const float* __restrict__ dinv,
                         const float* __restrict__ b1) {
    long long idx = (long long)blockIdx.x * blockDim.x + threadIdx.x;
    if (idx >= (long long)NN * HID) return;
    int i = (int)(idx >> 6);
    int f = (int)idx & 63;
    float di = dinv[i];
    agg1[idx] = leaky(agg1[idx] + xw1[idx] * di * di + b1[f]);
}

// xw2 = h1 @ W2  (64 -> 1): per-node dot product with float4 loads
__global__ void gemm2(const float* __restrict__ h1,
                      const float* __restrict__ W2,
                      float* __restrict__ xw2) {
    int i = blockIdx.x * blockDim.x + threadIdx.x;
    if (i >= NN) return;
    const float4* hr = (const float4*)(h1 + (size_t)i * HID);
    const float4* wr = (const float4*)W2;
    float acc = 0.0f;
    #pragma unroll
    for (int q = 0; q < HID / 4; ++q) {
        float4 h = hr[q], w = wr[q];
        acc += h.x * w.x + h.y * w.y + h.z * w.z + h.w * w.w;
    }
    xw2[i] = acc;
}

__global__ void edge_scatter2(const int* __restrict__ src,
                              const int* __restrict__ dst,
                              const float* __restrict__ edge_attr,
                              const float* __restrict__ dinv,
                              const float* __restrict__ xw2,
                              float* __restrict__ agg2) {
    int e = blockIdx.x * blockDim.x + threadIdx.x;
    if (e >= EE) return;
    int s = src[e], d = dst[e];
    float nw = dinv[s] * edge_attr[2 * e] * dinv[d];
    atomic_add_f32(&agg2[d], xw2[s] * nw);
}

// h2 = leaky(agg2 + xw2 * dinv^2 + b2), in place over xw2
__global__ void combine2(float* __restrict__ xw2,
                         const float* __restrict__ agg2,
                         const float* __restrict__ dinv,
                         const float* __restrict__ b2) {
    int i = blockIdx.x * blockDim.x + threadIdx.x;
    if (i >= NN) return;
    float di = dinv[i];
    xw2[i] = leaky(agg2[i] + xw2[i] * di * di + b2[0]);
}

// Per-graph head: FC1(4096->128) + leaky, FC2(128->2), softmax.
// h2 tile staged in LDS (16 KB of the 320 KB WGP LDS).
__global__ void head(const float* __restrict__ h2,
                     const float* __restrict__ fc1W,
                     const float* __restrict__ fc1b,
                     const float* __restrict__ fc2W,
                     const float* __restrict__ fc2b,
                     float* __restrict__ out) {
    __shared__ float sh[NODES_PER_G];
    __shared__ float mid[FC_HID];
    int g = blockIdx.x;
    int t = threadIdx.x;                       // 128 threads

    const float* hg = h2 + (size_t)g * NODES_PER_G;
    for (int i = t; i < NODES_PER_G; i += FC_HID) sh[i] = hg[i];
    __syncthreads();

    float acc = fc1b[t];
    #pragma unroll 8
    for (int i = 0; i < NODES_PER_G; ++i)
        acc += sh[i] * fc1W[(size_t)i * FC_HID + t];   // coalesced across t
    mid[t] = leaky(acc);
    __syncthreads();

    if (t == 0) {
        float o0 = fc2b[0], o1 = fc2b[1];
        #pragma unroll 8
        for (int j = 0; j < FC_HID; ++j) {
            float m = mid[j];
            o0 += m * fc2W[2 * j + 0];
            o1 += m * fc2W[2 * j + 1];
        }
        float mx = fmaxf(o0, o1);
        float e0 = __expf(o0 - mx), e1 = __expf(o1 - mx);
        float s  = e0 + e1;
        out[2 * g + 0] = e0 / s;
        out[2 * g + 1] = e1 / s;
    }
}

// ---------------- launch ----------------
extern "C" void kernel_launch(void* const* d_in, const int* in_sizes, int n_in,
                              void* d_out, int out_size, void* d_ws, size_t ws_size,
                              hipStream_t stream) {
    const float* x        = (const float*)d_in[0];
    const int*   eidx     = (const int*)d_in[1];     // [2, E]: src then dst
    const float* eattr    = (const float*)d_in[2];
    const float* W1       = (const float*)d_in[3];
    const float* b1       = (const float*)d_in[4];
    const float* W2       = (const float*)d_in[5];
    const float* b2       = (const float*)d_in[6];
    const float* fc1W     = (const float*)d_in[7];
    const float* fc1b     = (const float*)d_in[8];
    const float* fc2W     = (const float*)d_in[9];
    const float* fc2b     = (const float*)d_in[10];
    float*       out      = (float*)d_out;
    float*       ws       = (float*)d_ws;

    const int* src = eidx;
    const int* dst = eidx + EE;

    float* dinv = ws + OFF_DINV;
    float* agg1 = ws + OFF_AGG1;
    float* agg2 = ws + OFF_AGG2;
    float* xw1  = ws + OFF_XW1;
    float* xw2  = ws + OFF_XW2;

    const int B = 256;

    zero_ws<<<(ZERO_CNT + B - 1) / B, B, 0, stream>>>(ws);
    deg_scatter<<<(EE + B - 1) / B, B, 0, stream>>>(dst, eattr, dinv);
    dinv_finalize<<<(NN + B - 1) / B, B, 0, stream>>>(dinv);

    // 2048 waves, 4 waves (128 threads) per block
    gemm1_wmma<<<(NN / 16) / 4, 128, 0, stream>>>(x, W1, xw1);

    long long sc1 = (long long)EE * 16;
    edge_scatter1<<<(unsigned)((sc1 + B - 1) / B), B, 0, stream>>>(src, dst, eattr, dinv, xw1, agg1);

    long long cb1 = (long long)NN * HID;
    combine1<<<(unsigned)((cb1 + B - 1) / B), B, 0, stream>>>(agg1, xw1, dinv, b1);

    gemm2<<<(NN + B - 1) / B, B, 0, stream>>>(agg1 /*h1*/, W2, xw2);
    edge_scatter2<<<(EE + B - 1) / B, B, 0, stream>>>(src, dst, eattr, dinv, xw2, agg2);
    combine2<<<(NN + B - 1) / B, B, 0, stream>>>(xw2, agg2, dinv, b2);

    head<<<N_GRAPHS, FC_HID, 0, stream>>>(xw2 /*h2*/, fc1W, fc1b, fc2W, fc2b, out);
}